// DenseContrastiveLoss_27049704030937
// MI455X (gfx1250) — compile-verified
//
#include <hip/hip_runtime.h>
#include <hip/hip_bf16.h>
#include <math.h>

// ---------------------------------------------------------------------------
// DenseContrastiveLoss on MI455X (gfx1250, wave32).
// Heavy op: neg_sim[3136,65536] = (q_n @ queue^T) * (5 * rnorm_queue[col])
// done with V_WMMA_F32_16X16X4_F32. Everything else is bandwidth-trivial.
// ---------------------------------------------------------------------------

typedef __attribute__((ext_vector_type(2))) float v2f;
typedef __attribute__((ext_vector_type(8))) float v8f;

#define ROWS   3136      // B*N = 16*196
#define BATCH  16
#define NPIX   196
#define DDIM   128
#define QROWS  65536
#define INV_T  5.0f      // 1/0.2

__device__ __forceinline__ float waveSum(float v) {
    #pragma unroll
    for (int off = 16; off > 0; off >>= 1)
        v += __shfl_xor(v, off, 32);
    return v;
}

// ---- 1) L2-normalize rows of a [rows,128] matrix: one wave per row --------
__global__ void dcl_normalize_rows(const float* __restrict__ src,
                                   float* __restrict__ dst, int rows) {
    const int lane = threadIdx.x & 31;
    const int wave = threadIdx.x >> 5;
    const int row  = blockIdx.x * 8 + wave;
    if (row >= rows) return;
    float4 v = ((const float4*)src)[(size_t)row * 32 + lane];
    float ss = v.x * v.x + v.y * v.y + v.z * v.z + v.w * v.w;
    ss = waveSum(ss);
    float s = 1.0f / fmaxf(sqrtf(ss), 1e-12f);
    v.x *= s; v.y *= s; v.z *= s; v.w *= s;
    ((float4*)dst)[(size_t)row * 32 + lane] = v;
}

// ---- 2) Inverse L2 norms of queue rows: one wave per row ------------------
__global__ void dcl_queue_rnorm(const float* __restrict__ queue,
                                float* __restrict__ rq) {
    const int lane = threadIdx.x & 31;
    const int wave = threadIdx.x >> 5;
    const int row  = blockIdx.x * 8 + wave;
    float4 v = ((const float4*)queue)[(size_t)row * 32 + lane];
    float ss = v.x * v.x + v.y * v.y + v.z * v.z + v.w * v.w;
    ss = waveSum(ss);
    if (lane == 0) rq[row] = 1.0f / fmaxf(sqrtf(ss), 1e-12f);
}

// ---- 3) pos_sim[b,n] = max_m (q_n[b,n] . k_n[b,m]) / T --------------------
// (argmax-gathered positive similarity == the max itself)
__global__ void dcl_pos_sim(const float* __restrict__ qn,
                            const float* __restrict__ kn,
                            float* __restrict__ pos_out) {
    const int lane = threadIdx.x & 31;
    const int wave = threadIdx.x >> 5;
    const int r    = blockIdx.x * 8 + wave;       // 0..3135
    const int b    = r / NPIX;
    float4 q4 = ((const float4*)qn)[(size_t)r * 32 + lane];
    float best = -INFINITY;
    for (int m = 0; m < NPIX; ++m) {
        float4 k4 = ((const float4*)kn)[((size_t)b * NPIX + m) * 32 + lane];
        float d = q4.x * k4.x + q4.y * k4.y + q4.z * k4.z + q4.w * k4.w;
        d = waveSum(d);
        best = fmaxf(best, d);
    }
    if (lane == 0) pos_out[r] = best * INV_T;
}

// ---- 4) neg_sim GEMM via V_WMMA_F32_16X16X4_F32 ---------------------------
// Grid: (65536/512, 3136/16) ; block 256 = 8 waves; wave -> 16x64 tile.
// A frag (16x4): lane holds q_n[mb + (lane&15)][kk + 2*(lane>>4) + {0,1}]
// B frag (4x16): lane holds queue[nb + (lane&15)][kk + 2*(lane>>4) + {0,1}]
// C/D (16x16 f32): vgpr i, lanes0-15 -> M=i, lanes16-31 -> M=i+8.
// Column normalization + 1/T folded into the epilogue scale.
__global__ void dcl_neg_sim_wmma(const float* __restrict__ qn,
                                 const float* __restrict__ queue,
                                 const float* __restrict__ rq,
                                 float* __restrict__ neg_out) {
    const int lane = threadIdx.x & 31;
    const int wave = threadIdx.x >> 5;
    const int lm   = lane & 15;
    const int hi   = lane >> 4;

    const int    mb = blockIdx.y * 16;
    const size_t nb = (size_t)blockIdx.x * 512 + (size_t)wave * 64;

    const float* aBase = qn    + ((size_t)(mb + lm)) * DDIM + 2 * hi;
    const float* bBase = queue + (nb + lm) * DDIM + 2 * hi;

    v8f zero = {0.f, 0.f, 0.f, 0.f, 0.f, 0.f, 0.f, 0.f};
    v8f acc0 = zero, acc1 = zero, acc2 = zero, acc3 = zero;

    #pragma unroll
    for (int kk = 0; kk < DDIM; kk += 4) {
        v2f a  = *(const v2f*)(aBase + kk);
        v2f b0 = *(const v2f*)(bBase + kk);
        v2f b1 = *(const v2f*)(bBase + 16 * DDIM + kk);
        v2f b2 = *(const v2f*)(bBase + 32 * DDIM + kk);
        v2f b3 = *(const v2f*)(bBase + 48 * DDIM + kk);
        acc0 = __builtin_amdgcn_wmma_f32_16x16x4_f32(false, a, false, b0,
                                                     (short)0, acc0, false, false);
        acc1 = __builtin_amdgcn_wmma_f32_16x16x4_f32(false, a, false, b1,
                                                     (short)0, acc1, false, false);
        acc2 = __builtin_amdgcn_wmma_f32_16x16x4_f32(false, a, false, b2,
                                                     (short)0, acc2, false, false);
        acc3 = __builtin_amdgcn_wmma_f32_16x16x4_f32(false, a, false, b3,
                                                     (short)0, acc3, false, false);
    }

    // epilogue: scale by 5 * rnorm(queue col), store [row, col]
    const float s0 = INV_T * rq[nb + lm];
    const float s1 = INV_T * rq[nb + 16 + lm];
    const float s2 = INV_T * rq[nb + 32 + lm];
    const float s3 = INV_T * rq[nb + 48 + lm];
    #pragma unroll
    for (int i = 0; i < 8; ++i) {
        const size_t row = (size_t)(mb + i + 8 * hi);
        float* o = neg_out + row * (size_t)QROWS + nb + lm;
        o[0]  = acc0[i] * s0;
        o[16] = acc1[i] * s1;
        o[32] = acc2[i] * s2;
        o[48] = acc3[i] * s3;
    }
}

// ---- 5) per-row -log_softmax[0]: one block per row, fixed-order reduce ----
__global__ void dcl_row_loss(const float* __restrict__ out_full,
                             float* __restrict__ rowloss) {
    const int r = blockIdx.x;
    const int t = threadIdx.x;
    const float pos = out_full[1 + r];
    const float* neg = out_full + 1 + ROWS + (size_t)r * QROWS;

    float m = -INFINITY, s = 0.0f;
    for (int i = t; i < QROWS; i += 256) {
        float x = neg[i];
        if (x > m) { s = s * expf(m - x) + 1.0f; m = x; }
        else       { s += expf(x - m); }
    }
    __shared__ float sm[256];
    __shared__ float ss[256];
    sm[t] = m; ss[t] = s;
    __syncthreads();
    for (int off = 128; off > 0; off >>= 1) {
        if (t < off) {
            float m2 = sm[t + off], s2 = ss[t + off];
            float M = fmaxf(sm[t], m2);
            ss[t] = ss[t] * expf(sm[t] - M) + s2 * expf(m2 - M);
            sm[t] = M;
        }
        __syncthreads();
    }
    if (t == 0) {
        float M = fmaxf(sm[0], pos);
        float S = ss[0] * expf(sm[0] - M) + expf(pos - M);
        rowloss[r] = -(pos - M - logf(S));
    }
}

// ---- 6) mean over 3136 row losses: single block, fixed order --------------
__global__ void dcl_mean_loss(const float* __restrict__ rowloss,
                              float* __restrict__ out0) {
    const int t = threadIdx.x;
    float s = 0.0f;
    for (int i = t; i < ROWS; i += 256) s += rowloss[i];
    __shared__ float sh[256];
    sh[t] = s;
    __syncthreads();
    for (int off = 128; off > 0; off >>= 1) {
        if (t < off) sh[t] += sh[t + off];
        __syncthreads();
    }
    if (t == 0) out0[0] = sh[0] / (float)ROWS;
}

extern "C" void kernel_launch(void* const* d_in, const int* in_sizes, int n_in,
                              void* d_out, int out_size, void* d_ws, size_t ws_size,
                              hipStream_t stream) {
    const float* dense1 = (const float*)d_in[0];
    const float* dense2 = (const float*)d_in[1];
    // d_in[2], d_in[3] (backbone features) are unused by the reference.
    const float* queue  = (const float*)d_in[4];
    float* out = (float*)d_out;

    // workspace layout (< 4 MB)
    float* qn      = (float*)d_ws;              // 3136*128
    float* kn      = qn + (size_t)ROWS * DDIM;  // 3136*128
    float* rq      = kn + (size_t)ROWS * DDIM;  // 65536 inverse queue norms
    float* rowloss = rq + QROWS;                // 3136

    float* pos_out = out + 1;
    float* neg_out = out + 1 + ROWS;

    dcl_normalize_rows<<<ROWS / 8, 256, 0, stream>>>(dense1, qn, ROWS);
    dcl_normalize_rows<<<ROWS / 8, 256, 0, stream>>>(dense2, kn, ROWS);
    dcl_queue_rnorm<<<QROWS / 8, 256, 0, stream>>>(queue, rq);
    dcl_pos_sim<<<ROWS / 8, 256, 0, stream>>>(qn, kn, pos_out);

    dim3 grid(QROWS / 512, ROWS / 16);   // 128 x 196 workgroups
    dcl_neg_sim_wmma<<<grid, 256, 0, stream>>>(qn, queue, rq, neg_out);

    dcl_row_loss<<<ROWS, 256, 0, stream>>>(out, rowloss);
    dcl_mean_loss<<<1, 256, 0, stream>>>(rowloss, out);
}